// VARLoss_24026047054559
// MI455X (gfx1250) — compile-verified
//
#include <hip/hip_runtime.h>
#include <hip/hip_bf16.h>
#include <stdint.h>

// Problem constants (from reference): x[B, T, D], B=4096, T=256, D=64
#define BN   4096
#define TN   256
#define DN   64
#define KSEL 204        // int(0.05 * 4096)
#define COLS 16         // d-columns per block (tile width)
#define EPSV 1e-8f

typedef uint32_t u32;
typedef uint64_t u64;
typedef __attribute__((ext_vector_type(4))) u32 v4u;
typedef __attribute__((ext_vector_type(8))) u32 v8u;

// LDS layout (dynamic shared memory, starts at LDS offset 0):
//   [0, 256KB)           : float data[4096][16]   (TDM destination)
//   [256KB, 256KB+16KB)  : u32 hist[256][16]      (bin-major -> bank = col)
//   then                 : u32 prefix[16], target[16]
#define DATA_BYTES (BN * COLS * 4)          // 262144
#define HIST_BYTES (256 * COLS * 4)         // 16384
#define STATE_BYTES (2 * COLS * 4)
#define DYN_LDS (DATA_BYTES + HIST_BYTES + STATE_BYTES)

__global__ void var_select_kernel(const float* __restrict__ x_fake,
                                  const float* __restrict__ x_real,
                                  float* __restrict__ var_out /* [2][T*D] */) {
    extern __shared__ unsigned char smem[];
    float* data  = (float*)smem;                              // [4096][16]
    u32*   hist  = (u32*)(smem + DATA_BYTES);                 // [256][16]
    u32*   pref  = (u32*)(smem + DATA_BYTES + HIST_BYTES);    // [16]
    u32*   targ  = pref + COLS;                               // [16]

    const u32 bid  = blockIdx.x;
    const u32 arr  = bid & 1u;          // 0 = fake, 1 = real
    const u32 rest = bid >> 1;
    const u32 t    = rest >> 2;         // 0..255
    const u32 dg   = rest & 3u;         // 0..3
    const u32 d0   = dg * COLS;

    const float* src = arr ? x_real : x_fake;

    // ---- TDM: load the strided column tile into LDS (wave 0 only) ----
    if (threadIdx.x < 32) {
        u64 ga = (u64)(uintptr_t)src + 4ull * ((u64)t * DN + d0);

        v4u g0;
        g0.x = 1u;                                        // count=1, user descriptor
        g0.y = 0u;                                        // lds_addr = 0 (dyn LDS base)
        g0.z = (u32)(ga & 0xFFFFFFFFull);                 // global_addr[31:0]
        g0.w = (u32)((ga >> 32) & 0x1FFFFFFull)           // global_addr[56:32]
             | (2u << 30);                                // type = 2 ("image")

        v8u g1;
        g1[0] = (2u << 16);                               // data_size = 4 bytes
        g1[1] = ((u32)COLS << 16);                        // tensor_dim0 = 16 (lo half)
        g1[2] = ((u32)BN << 16);                          // dim0 hi = 0; tensor_dim1 lo = 4096
        g1[3] = ((u32)COLS << 16);                        // dim1 hi = 0; tile_dim0 = 16
        g1[4] = (u32)BN;                                  // tile_dim1 = 4096; tile_dim2 = 0
        g1[5] = (u32)(TN * DN);                           // tensor_dim0_stride = 16384 elems
        g1[6] = 0u;
        g1[7] = 0u;

        asm volatile("tensor_load_to_lds %0, %1"
                     :: "s"(g0), "s"(g1)
                     : "memory");
        __builtin_amdgcn_s_wait_tensorcnt(0);
    }
    if (threadIdx.x < COLS) {
        pref[threadIdx.x] = 0u;
        targ[threadIdx.x] = KSEL;
    }
    __syncthreads();

    const u32 col = threadIdx.x & (COLS - 1);   // 0..15
    const u32 row = threadIdx.x >> 4;           // 0..15 (each handles 256 b-values)

    // ---- exact radix select (MSB-first, 4 x 8-bit digits) ----
    for (int pass = 0; pass < 4; ++pass) {
        // zero histogram
        for (u32 i = threadIdx.x; i < 256u * COLS; i += blockDim.x) hist[i] = 0u;
        __syncthreads();

        const u32 shift = 24u - 8u * (u32)pass;
        const u64 pfx   = (u64)pref[col];

        for (u32 j = 0; j < BN / COLS; ++j) {
            u32 b = row + COLS * j;                       // conflict-free LDS read
            u32 u = __float_as_uint(data[b * COLS + col]);
            u32 key = u ^ ((u >> 31) ? 0xFFFFFFFFu : 0x80000000u);
            // prefix match (shift+8 may be 32 -> do it in 64-bit)
            if ((((u64)key) >> (shift + 8u)) == pfx) {
                u32 bin = (key >> shift) & 0xFFu;
                atomicAdd(&hist[bin * COLS + col], 1u);
            }
        }
        __syncthreads();

        // per-column serial scan of 256 bins (16 columns in parallel)
        if (threadIdx.x < COLS) {
            u32 c = threadIdx.x;
            u32 tgt = targ[c];
            u32 cum = 0u, dsel = 0u;
            for (u32 bin = 0; bin < 256u; ++bin) {
                u32 h = hist[bin * COLS + c];
                if (cum + h > tgt) { dsel = bin; break; }
                cum += h;
            }
            targ[c] = tgt - cum;
            pref[c] = (pref[c] << 8) | dsel;
        }
        __syncthreads();
    }

    if (threadIdx.x < COLS) {
        u32 key = pref[threadIdx.x];   // full 32-bit ordered key of k-th smallest
        u32 u   = (key & 0x80000000u) ? (key ^ 0x80000000u) : ~key;
        var_out[arr * (TN * DN) + t * DN + d0 + threadIdx.x] = __uint_as_float(u);
    }
}

__global__ void var_reduce_kernel(const float* __restrict__ var_out,
                                  float* __restrict__ out) {
    __shared__ float s_abs[256];
    __shared__ float s_rel[256];
    float acc_a = 0.f, acc_r = 0.f;
    for (u32 i = threadIdx.x; i < (u32)(TN * DN); i += blockDim.x) {
        float vf = var_out[i];
        float vr = var_out[TN * DN + i];
        float d  = fabsf(vf - vr);
        acc_a += d;
        acc_r += d / (fabsf(vr) + EPSV);
    }
    s_abs[threadIdx.x] = acc_a;
    s_rel[threadIdx.x] = acc_r;
    __syncthreads();
    for (u32 s = 128; s > 0; s >>= 1) {
        if (threadIdx.x < s) {
            s_abs[threadIdx.x] += s_abs[threadIdx.x + s];
            s_rel[threadIdx.x] += s_rel[threadIdx.x + s];
        }
        __syncthreads();
    }
    if (threadIdx.x == 0) {
        const float inv = 1.0f / (float)(TN * DN);
        out[0] = s_abs[0] * inv;
        out[1] = s_rel[0] * inv;
    }
}

extern "C" void kernel_launch(void* const* d_in, const int* in_sizes, int n_in,
                              void* d_out, int out_size, void* d_ws, size_t ws_size,
                              hipStream_t stream) {
    (void)in_sizes; (void)n_in; (void)out_size; (void)ws_size;
    const float* x_fake = (const float*)d_in[0];
    const float* x_real = (const float*)d_in[1];
    float* var_ws = (float*)d_ws;       // 2 * 16384 floats
    float* out    = (float*)d_out;      // 2 floats

    // Allow > 64 KB of dynamic LDS (kernel uses ~272 KB of the 320 KB WGP LDS).
    (void)hipFuncSetAttribute((const void*)var_select_kernel,
                              hipFuncAttributeMaxDynamicSharedMemorySize,
                              DYN_LDS);

    const int grid = TN * (DN / COLS) * 2;   // 2048 blocks
    var_select_kernel<<<grid, 256, DYN_LDS, stream>>>(x_fake, x_real, var_ws);
    var_reduce_kernel<<<1, 256, 0, stream>>>(var_ws, out);
}